// TopologicalDynamics_18932215841384
// MI455X (gfx1250) — compile-verified
//
#include <hip/hip_runtime.h>
#include <stdint.h>

// Problem geometry (fixed by the reference: psi is 32 x 1024 x 1024 fp32).
#define W 1024
#define H 1024

constexpr int   TX    = 128;        // outputs per block in x
constexpr int   TY    = 32;         // outputs per block in y
constexpr int   LDSW  = 136;        // floats per LDS row = 34 float4 (x halo: -4..+131)
constexpr int   LDSH  = TY + 2;     // 34 rows (y halo: -1..+32)
constexpr float KC    = 0.002f;     // DT * (GAMMA + BETA) = 0.01 * 0.2

__global__ __launch_bounds__(256)
void topo_stencil(const float* __restrict__ psi, float* __restrict__ out) {
    const int z   = blockIdx.z;
    const int gx0 = blockIdx.x * TX;
    const int gy0 = blockIdx.y * TY;
    const float* slice = psi + (size_t)z * (size_t)(W * H);

    __shared__ float tile[LDSH * LDSW];

    // ---------------------------------------------------------------
    // Stage (34 rows x 136 cols) = tile + halo from HBM into LDS with
    // gfx1250 async global->LDS b128 copies (ASYNCcnt path, no VGPR data).
    // Halo addresses are clamped into the slice (no OOB faults); clamped
    // cells hold garbage that is masked to zero during compute below,
    // which reproduces the reference conv's zero padding.
    // ---------------------------------------------------------------
    const int tid = threadIdx.x;
    for (int i = tid; i < LDSH * (LDSW / 4); i += 256) {
        const int r  = i / (LDSW / 4);
        const int c4 = i % (LDSW / 4);
        int gy = gy0 - 1 + r;
        gy = gy < 0 ? 0 : (gy > H - 1 ? H - 1 : gy);
        int gx = gx0 - 4 + c4 * 4;
        gx = gx < 0 ? 0 : (gx > W - 4 ? W - 4 : gx);
        const float* gp = slice + (size_t)gy * W + gx;
        // Low 32 bits of a generic pointer to LDS are the LDS byte offset
        // (flat-aperture layout: addr[31:0] = LDS offset).
        const uint32_t lds_off = (uint32_t)(uintptr_t)(&tile[r * LDSW + c4 * 4]);
        asm volatile("global_load_async_to_lds_b128 %0, %1, off"
                     :: "v"(lds_off), "v"((uint64_t)(uintptr_t)gp)
                     : "memory");
    }
    // Each wave drains its own async copies, then the barrier publishes LDS.
    asm volatile("s_wait_asynccnt 0" ::: "memory");
    __syncthreads();

    // ---------------------------------------------------------------
    // 5-point stencil from LDS: each thread computes 4 float4 outputs.
    // lanes 0..31 of a wave cover one 128-wide row (32 consecutive float4),
    // 8 row-groups x 4 iterations cover the 32-row tile.
    // ---------------------------------------------------------------
    const int lane4 = tid & 31;     // float4 column within the tile
    const int ly0   = tid >> 5;     // 0..7
    const int lx    = lane4 * 4;
    const int gx    = gx0 + lx;

    float* oslice = out + (size_t)z * (size_t)(W * H);

#pragma unroll
    for (int k = 0; k < 4; ++k) {
        const int ly  = ly0 + k * 8;
        const int gy  = gy0 + ly;
        const int row = ly + 1;
        const int col = lx + 4;

        const float4 c = *(const float4*)&tile[row * LDSW + col];
        float4 u = *(const float4*)&tile[(row - 1) * LDSW + col];
        float4 d = *(const float4*)&tile[(row + 1) * LDSW + col];
        float  l = tile[row * LDSW + col - 1];
        float  r = tile[row * LDSW + col + 4];

        // Zero-padding boundary conditions (mask clamped halo garbage).
        if (gy == 0)     { u.x = u.y = u.z = u.w = 0.0f; }
        if (gy == H - 1) { d.x = d.y = d.z = d.w = 0.0f; }
        if (gx == 0)       l = 0.0f;
        if (gx + 4 == W)   r = 0.0f;

        float4 o;
        o.x = c.x + KC * (u.x + d.x + l   + c.y - 4.0f * c.x);
        o.y = c.y + KC * (u.y + d.y + c.x + c.z - 4.0f * c.y);
        o.z = c.z + KC * (u.z + d.z + c.y + c.w - 4.0f * c.z);
        o.w = c.w + KC * (u.w + d.w + c.z + r   - 4.0f * c.w);

        *(float4*)&oslice[(size_t)gy * W + gx] = o;
    }
}

extern "C" void kernel_launch(void* const* d_in, const int* in_sizes, int n_in,
                              void* d_out, int out_size, void* d_ws, size_t ws_size,
                              hipStream_t stream) {
    (void)n_in; (void)out_size; (void)d_ws; (void)ws_size;
    const float* psi = (const float*)d_in[0];
    float*       out = (float*)d_out;
    const int Z = in_sizes[0] / (W * H);   // 32 for the reference shapes
    dim3 grid(W / TX, H / TY, Z);
    topo_stencil<<<grid, dim3(256, 1, 1), 0, stream>>>(psi, out);
}